// FCHCGNN_10385230922560
// MI455X (gfx1250) — compile-verified
//
#include <hip/hip_runtime.h>
#include <hip/hip_bf16.h>

// ---------------------------------------------------------------------------
// 3-layer GCN for MI455X (gfx1250, wave32).
//   per layer:  h = X @ W   (WMMA f32 16x16x4, W staged in LDS)
//               agg = D^-1/2 (A+I) D^-1/2 h   (self-init + edge scatter atomics)
//               out = act(agg + b)
//   final: log_softmax over 64 dims (wave-per-node shuffle reduction)
// ---------------------------------------------------------------------------

typedef __attribute__((ext_vector_type(2))) float v2f;
typedef __attribute__((ext_vector_type(8))) float v8f;

#define WAVES_PER_BLOCK 8

// ---------------------------- degree / norm --------------------------------

__global__ void deg_init_kernel(float* __restrict__ deg, int n) {
    int i = blockIdx.x * blockDim.x + threadIdx.x;
    if (i < n) deg[i] = 1.0f;                 // self-loop contributes 1
}

__global__ void deg_count_kernel(const int* __restrict__ dst,
                                 float* __restrict__ deg, int e) {
    int i = blockIdx.x * blockDim.x + threadIdx.x;
    if (i < e) atomicAdd(&deg[dst[i]], 1.0f);
}

__global__ void deg_rsqrt_kernel(float* __restrict__ deg, int n) {
    int i = blockIdx.x * blockDim.x + threadIdx.x;
    if (i < n) deg[i] = rsqrtf(deg[i]);       // deg >= 1 always (self-loop)
}

// ------------------------------ WMMA GEMM ----------------------------------
// C[M x FOUT] = A[M x K] * W[K x FOUT]; M multiple of 16; K, FOUT mult of 16.
// Each wave computes a 16 x (16*NT) strip: one A fragment per K-step feeds NT
// independent V_WMMA_F32_16X16X4_F32 chains (breaks the accumulator RAW chain
// and divides redundant A traffic by NT). W lives in LDS (<= 64KB).
//
// f32 A (16x4) lane layout (ISA 7.12.2): lanes 0-15 row=lane,   K = k0+{0,1};
//                                        lanes 16-31 row=ln-16, K = k0+{2,3}.
// f32 B (4x16): lanes 0-15 col=lane, K=k0+{0,1}; lanes 16-31 col=ln-16, K=k0+{2,3}.
// f32 C/D (16x16): VGPR r holds row (r + 8*(lane>>4)), col = lane&15.

template <int K, int FOUT, int NT>
__global__ __launch_bounds__(256) void gcn_gemm_kernel(
    const float* __restrict__ A, const float* __restrict__ W,
    float* __restrict__ C, int M) {
    constexpr int GN = FOUT / (16 * NT);      // N-strips per 16-row block
    __shared__ float sW[K * FOUT];

    // cooperative 128-bit staging of W into LDS
    const int nvec = (K * FOUT) / 4;
    for (int i = threadIdx.x; i < nvec; i += blockDim.x)
        ((float4*)sW)[i] = ((const float4*)W)[i];
    __syncthreads();

    const int wave  = threadIdx.x >> 5;
    const int lane  = threadIdx.x & 31;
    const int wid   = blockIdx.x * WAVES_PER_BLOCK + wave;
    const int tileM = wid / GN;
    if (tileM * 16 >= M) return;              // wave-uniform guard (EXEC all-ones)
    const int nbase = (wid % GN) * (16 * NT);
    const int row   = tileM * 16 + (lane & 15);
    const int khalf = (lane >> 4) * 2;        // 0 for lanes 0-15, 2 for 16-31

    v8f acc[NT];
#pragma unroll
    for (int t = 0; t < NT; ++t) acc[t] = (v8f){};

#pragma unroll
    for (int k0 = 0; k0 < K; k0 += 4) {
        v2f a;
        const float* ap = A + (size_t)row * K + k0 + khalf;
        a.x = ap[0];
        a.y = ap[1];
        const float* bp = sW + (k0 + khalf) * FOUT + nbase + (lane & 15);
#pragma unroll
        for (int t = 0; t < NT; ++t) {
            v2f b;
            b.x = bp[t * 16];
            b.y = bp[t * 16 + FOUT];
            acc[t] = __builtin_amdgcn_wmma_f32_16x16x4_f32(
                /*neg_a=*/false, a, /*neg_b=*/false, b,
                /*c_mod=*/(short)0, acc[t], /*reuse_a=*/false, /*reuse_b=*/false);
        }
    }

    const int rbase = tileM * 16 + (lane >> 4) * 8;
#pragma unroll
    for (int t = 0; t < NT; ++t) {
        float* cp = C + (size_t)rbase * FOUT + nbase + t * 16 + (lane & 15);
#pragma unroll
        for (int r = 0; r < 8; ++r) cp[(size_t)r * FOUT] = acc[t][r];
    }
}

// -------------------------- aggregation stages -----------------------------

// out[i,:] = h[i,:] * dis[i]^2   (self-loop term; also zero-initializes out)
template <int F>
__global__ void gcn_selfinit_kernel(const float* __restrict__ h,
                                    const float* __restrict__ dis,
                                    float* __restrict__ out, int n) {
    constexpr int C4 = F / 4;
    unsigned i = blockIdx.x * blockDim.x + threadIdx.x;   // float4 index
    if (i >= (unsigned)n * C4) return;
    float d  = dis[i / C4];
    float dd = d * d;
    float4 v = ((const float4*)h)[i];
    v.x *= dd; v.y *= dd; v.z *= dd; v.w *= dd;
    ((float4*)out)[i] = v;
}

// one wave per edge: out[dst,:] += h[src,:] * dis[src]*dis[dst]
template <int F>
__global__ void gcn_scatter_kernel(const float* __restrict__ h,
                                   const int* __restrict__ src,
                                   const int* __restrict__ dst,
                                   const float* __restrict__ dis,
                                   float* __restrict__ out, int e) {
    int edge = blockIdx.x * (blockDim.x >> 5) + (threadIdx.x >> 5);
    if (edge >= e) return;
    int lane = threadIdx.x & 31;
    int s = src[edge];
    int d = dst[edge];
    float nrm = dis[s] * dis[d];
    const float* hp = h + (size_t)s * F;
    float* op = out + (size_t)d * F;
#pragma unroll
    for (int f = lane; f < F; f += 32) atomicAdd(&op[f], hp[f] * nrm);
}

template <int F, bool RELU>
__global__ void gcn_bias_act_kernel(float* __restrict__ out,
                                    const float* __restrict__ b, int n) {
    constexpr int C4 = F / 4;
    unsigned i = blockIdx.x * blockDim.x + threadIdx.x;   // float4 index
    if (i >= (unsigned)n * C4) return;
    float4 v  = ((float4*)out)[i];
    float4 bb = ((const float4*)b)[i % C4];
    v.x += bb.x; v.y += bb.y; v.z += bb.z; v.w += bb.w;
    if (RELU) {
        v.x = fmaxf(v.x, 0.0f); v.y = fmaxf(v.y, 0.0f);
        v.z = fmaxf(v.z, 0.0f); v.w = fmaxf(v.w, 0.0f);
    }
    ((float4*)out)[i] = v;
}

// wave-per-node log_softmax over 64 features (2 per lane)
__global__ void gcn_logsoftmax_kernel(float* __restrict__ out, int n) {
    int node = blockIdx.x * (blockDim.x >> 5) + (threadIdx.x >> 5);
    if (node >= n) return;
    int lane = threadIdx.x & 31;
    float* p = out + (size_t)node * 64;
    float v0 = p[lane];
    float v1 = p[lane + 32];
    float m = fmaxf(v0, v1);
#pragma unroll
    for (int off = 16; off; off >>= 1) m = fmaxf(m, __shfl_xor(m, off, 32));
    float s = __expf(v0 - m) + __expf(v1 - m);
#pragma unroll
    for (int off = 16; off; off >>= 1) s += __shfl_xor(s, off, 32);
    float ls = __logf(s);
    p[lane]      = v0 - m - ls;
    p[lane + 32] = v1 - m - ls;
}

// ------------------------------- driver ------------------------------------

extern "C" void kernel_launch(void* const* d_in, const int* in_sizes, int n_in,
                              void* d_out, int out_size, void* d_ws, size_t ws_size,
                              hipStream_t stream) {
    const float* x  = (const float*)d_in[0];
    const int*   ei = (const int*)d_in[1];
    const float* W0 = (const float*)d_in[2];
    const float* b0 = (const float*)d_in[3];
    const float* W1 = (const float*)d_in[4];
    const float* b1 = (const float*)d_in[5];
    const float* W2 = (const float*)d_in[6];
    const float* b2 = (const float*)d_in[7];

    const int N = in_sizes[0] / 128;   // 100000 nodes
    const int E = in_sizes[1] / 2;     // 1.6M edges
    const int* src = ei;
    const int* dst = ei + E;

    char* ws = (char*)d_ws;
    float* dis  = (float*)ws;                                   // N floats
    float* hbuf = (float*)(ws + (1u << 19));                    // N*128 floats
    float* fbuf = (float*)(ws + (1u << 19) + (size_t)N * 128 * 4);
    float* outf = (float*)d_out;                                // N*64 floats

    const int T = 256;
    const unsigned nv128 = (unsigned)N * (128 / 4);   // float4 elements
    const unsigned nv64  = (unsigned)N * (64 / 4);

    // ---- symmetric normalization -------------------------------------------
    deg_init_kernel<<<(N + T - 1) / T, T, 0, stream>>>(dis, N);
    deg_count_kernel<<<(E + T - 1) / T, T, 0, stream>>>(dst, dis, E);
    deg_rsqrt_kernel<<<(N + T - 1) / T, T, 0, stream>>>(dis, N);

    const int mtiles = N / 16;                 // 6250, exact
    const int eblocks = (E + WAVES_PER_BLOCK - 1) / WAVES_PER_BLOCK;

    // ---- layer 0: x -> fbuf ------------------------------------------------
    {
        const int tiles = mtiles * 2;          // GN = 128/(16*4) = 2
        gcn_gemm_kernel<128, 128, 4><<<(tiles + WAVES_PER_BLOCK - 1) / WAVES_PER_BLOCK,
                                       T, 0, stream>>>(x, W0, hbuf, N);
        gcn_selfinit_kernel<128><<<(nv128 + T - 1) / T, T, 0, stream>>>(hbuf, dis, fbuf, N);
        gcn_scatter_kernel<128><<<eblocks, T, 0, stream>>>(hbuf, src, dst, dis, fbuf, E);
        gcn_bias_act_kernel<128, true><<<(nv128 + T - 1) / T, T, 0, stream>>>(fbuf, b0, N);
    }

    // ---- layer 1: fbuf -> fbuf (hbuf as GEMM scratch) ----------------------
    {
        const int tiles = mtiles * 2;
        gcn_gemm_kernel<128, 128, 4><<<(tiles + WAVES_PER_BLOCK - 1) / WAVES_PER_BLOCK,
                                       T, 0, stream>>>(fbuf, W1, hbuf, N);
        // fbuf fully consumed by the GEMM above; safe to reuse as aggregate dst
        gcn_selfinit_kernel<128><<<(nv128 + T - 1) / T, T, 0, stream>>>(hbuf, dis, fbuf, N);
        gcn_scatter_kernel<128><<<eblocks, T, 0, stream>>>(hbuf, src, dst, dis, fbuf, E);
        gcn_bias_act_kernel<128, true><<<(nv128 + T - 1) / T, T, 0, stream>>>(fbuf, b1, N);
    }

    // ---- layer 2: fbuf -> d_out (64 features) ------------------------------
    {
        const int tiles = mtiles;              // GN = 64/(16*4) = 1
        gcn_gemm_kernel<128, 64, 4><<<(tiles + WAVES_PER_BLOCK - 1) / WAVES_PER_BLOCK,
                                      T, 0, stream>>>(fbuf, W2, hbuf, N);
        gcn_selfinit_kernel<64><<<(nv64 + T - 1) / T, T, 0, stream>>>(hbuf, dis, outf, N);
        gcn_scatter_kernel<64><<<eblocks, T, 0, stream>>>(hbuf, src, dst, dis, outf, E);
        gcn_bias_act_kernel<64, false><<<(nv64 + T - 1) / T, T, 0, stream>>>(outf, b2, N);
        gcn_logsoftmax_kernel<<<(N + WAVES_PER_BLOCK - 1) / WAVES_PER_BLOCK,
                                T, 0, stream>>>(outf, N);
    }
}